// GATLayer_65163243815808
// MI455X (gfx1250) — compile-verified
//
#include <hip/hip_runtime.h>
#include <hip/hip_bf16.h>

typedef float v2f __attribute__((ext_vector_type(2)));
typedef float v8f __attribute__((ext_vector_type(8)));

#define IN_DIM 128
#define OUT_DIM 64

// ---- order-preserving float <-> uint encoding (for atomic float max) ----
__device__ __forceinline__ unsigned f2ord(float f) {
    unsigned b = __float_as_uint(f);
    return (b & 0x80000000u) ? ~b : (b | 0x80000000u);
}
__device__ __forceinline__ float ord2f(unsigned u) {
    unsigned b = (u & 0x80000000u) ? (u & 0x7FFFFFFFu) : ~u;
    return __uint_as_float(b);
}

// ---- init: zero output accumulator, denominators, max-encodings ----
__global__ void k_init(float* __restrict__ out, float* __restrict__ den,
                       unsigned* __restrict__ menc, int n, long total) {
    long i = (long)blockIdx.x * blockDim.x + threadIdx.x;
    if (i < total) out[i] = 0.0f;
    if (i < n) { den[i] = 0.0f; menc[i] = 0u; }
}

// ---- Wh = h @ W + b via V_WMMA_F32_16X16X4_F32 -------------------------
// One wave computes a 16-row x 64-col stripe: 4 accumulators (N-tiles),
// A fragment loaded once per K-step and reused for all 4 WMMAs.
__global__ void k_gemm_wmma(const float* __restrict__ h, const float* __restrict__ W,
                            const float* __restrict__ Wb, float* __restrict__ Wh, int n) {
    const int lane  = threadIdx.x & 31;
    const int wave  = threadIdx.x >> 5;
    const int tilesM = (n + 15) >> 4;
    const int mTile  = blockIdx.x * (blockDim.x >> 5) + wave;
    if (mTile >= tilesM) return;               // wave-uniform: EXEC stays all-1s
    const int mBase = mTile << 4;

    const int mrow = lane & 15;                // A: M = lane%16
    const int kk   = (lane >> 4) << 1;         // A: K pair offset 0 or 2
    int arowIdx = mBase + mrow; if (arowIdx > n - 1) arowIdx = n - 1;
    const float* __restrict__ arow = h + (size_t)arowIdx * IN_DIM;

    v8f acc0 = {}, acc1 = {}, acc2 = {}, acc3 = {};
    for (int k0 = 0; k0 < IN_DIM; k0 += 4) {
        v2f a;
        a.x = arow[k0 + kk];
        a.y = arow[k0 + kk + 1];
        const float* __restrict__ w0 = W + (size_t)(k0 + kk) * OUT_DIM;
        const float* __restrict__ w1 = w0 + OUT_DIM;
        v2f b0; b0.x = w0[mrow];      b0.y = w1[mrow];
        v2f b1; b1.x = w0[mrow + 16]; b1.y = w1[mrow + 16];
        v2f b2; b2.x = w0[mrow + 32]; b2.y = w1[mrow + 32];
        v2f b3; b3.x = w0[mrow + 48]; b3.y = w1[mrow + 48];
        acc0 = __builtin_amdgcn_wmma_f32_16x16x4_f32(false, a, false, b0, (short)0, acc0, false, false);
        acc1 = __builtin_amdgcn_wmma_f32_16x16x4_f32(false, a, false, b1, (short)0, acc1, false, false);
        acc2 = __builtin_amdgcn_wmma_f32_16x16x4_f32(false, a, false, b2, (short)0, acc2, false, false);
        acc3 = __builtin_amdgcn_wmma_f32_16x16x4_f32(false, a, false, b3, (short)0, acc3, false, false);
    }

    // C/D layout: VGPR r -> M = r (lanes 0-15) or r+8 (lanes 16-31), N = lane%16
    const int col  = lane & 15;
    const int rOff = (lane < 16) ? 0 : 8;
    const float bb0 = Wb[col], bb1 = Wb[col + 16], bb2 = Wb[col + 32], bb3 = Wb[col + 48];
    #pragma unroll
    for (int r = 0; r < 8; ++r) {
        int orow = mBase + rOff + r;
        if (orow < n) {
            float* o = Wh + (size_t)orow * OUT_DIM + col;
            o[0]  = acc0[r] + bb0;
            o[16] = acc1[r] + bb1;
            o[32] = acc2[r] + bb2;
            o[48] = acc3[r] + bb3;
        }
    }
}

// ---- per-node attention partials: a_dst = Wh . Aw[:64], a_src = Wh . Aw[64:] ----
// One wave per node; lane j handles columns j and j+32; wave32 shuffle reduction.
__global__ void k_node_att(const float* __restrict__ Wh, const float* __restrict__ Aw,
                           float* __restrict__ a_dst, float* __restrict__ a_src, int n) {
    const int lane = threadIdx.x & 31;
    const int wave = threadIdx.x >> 5;
    const int node = blockIdx.x * (blockDim.x >> 5) + wave;
    if (node >= n) return;
    const float* __restrict__ r = Wh + (size_t)node * OUT_DIM;
    float w0 = r[lane], w1 = r[lane + 32];
    float sd = w0 * Aw[lane]           + w1 * Aw[lane + 32];
    float ss = w0 * Aw[OUT_DIM + lane] + w1 * Aw[OUT_DIM + lane + 32];
    #pragma unroll
    for (int off = 16; off > 0; off >>= 1) {
        sd += __shfl_down(sd, off, 32);
        ss += __shfl_down(ss, off, 32);
    }
    if (lane == 0) { a_dst[node] = sd; a_src[node] = ss; }
}

// ---- per-edge logit + segment max (atomic u32 max on order-encoded float) ----
__global__ void k_edge_logit(const int* __restrict__ src, const int* __restrict__ dst,
                             const float* __restrict__ a_dst, const float* __restrict__ a_src,
                             const float* __restrict__ Ab, float* __restrict__ ebuf,
                             unsigned* __restrict__ menc, int E) {
    int i = blockIdx.x * blockDim.x + threadIdx.x;
    if (i >= E) return;
    int d = dst[i], s = src[i];
    float e = a_dst[d] + a_src[s] + Ab[0];
    e = (e > 0.0f) ? e : 0.2f * e;             // leaky_relu(0.2)
    ebuf[i] = e;
    atomicMax(&menc[d], f2ord(e));
}

// ---- decode maxes; isolated nodes (enc==0 / non-finite) -> 0, like reference ----
__global__ void k_fix_m(unsigned* __restrict__ menc, float* __restrict__ m, int n) {
    int i = blockIdx.x * blockDim.x + threadIdx.x;
    if (i >= n) return;
    unsigned u = menc[i];
    float v = (u == 0u) ? 0.0f : ord2f(u);
    if (!__builtin_isfinite(v)) v = 0.0f;
    m[i] = v;                                  // in-place reinterpret: same slot
}

// ---- per-edge exp + segment sum ----
__global__ void k_edge_exp(const int* __restrict__ dst, const float* __restrict__ m,
                           float* __restrict__ ebuf, float* __restrict__ den, int E) {
    int i = blockIdx.x * blockDim.x + threadIdx.x;
    if (i >= E) return;
    int d = dst[i];
    float ex = __expf(ebuf[i] - m[d]);
    ebuf[i] = ex;
    atomicAdd(&den[d], ex);
}

// ---- weighted scatter-sum: out[dst] += alpha * Wh[src] ----
// One wave per edge: contiguous 256B gather of Wh[src], 256B of fp32 atomics.
__global__ void k_aggregate(const int* __restrict__ src, const int* __restrict__ dst,
                            const float* __restrict__ ebuf, const float* __restrict__ den,
                            const float* __restrict__ Wh, float* __restrict__ out, int E) {
    const int lane = threadIdx.x & 31;
    const int wave = threadIdx.x >> 5;
    long e = (long)blockIdx.x * (blockDim.x >> 5) + wave;
    if (e >= E) return;
    int d = dst[e], s = src[e];
    float alpha = ebuf[e] / den[d];
    const float* __restrict__ wr = Wh + (size_t)s * OUT_DIM;
    float* __restrict__ orow = out + (size_t)d * OUT_DIM;
    atomicAdd(&orow[lane],      alpha * wr[lane]);
    atomicAdd(&orow[lane + 32], alpha * wr[lane + 32]);
}

extern "C" void kernel_launch(void* const* d_in, const int* in_sizes, int n_in,
                              void* d_out, int out_size, void* d_ws, size_t ws_size,
                              hipStream_t stream) {
    const float* h   = (const float*)d_in[0];
    const float* W_w = (const float*)d_in[1];
    const float* W_b = (const float*)d_in[2];
    const float* A_w = (const float*)d_in[3];
    const float* A_b = (const float*)d_in[4];
    const int*   src = (const int*)d_in[5];
    const int*   dst = (const int*)d_in[6];
    float* out = (float*)d_out;

    const int n = in_sizes[0] / IN_DIM;   // 50000
    const int E = in_sizes[5];            // 800000

    // workspace layout (floats)
    float* ws    = (float*)d_ws;
    float* Wh    = ws;                           // n*64
    float* a_dst = Wh + (size_t)n * OUT_DIM;     // n
    float* a_src = a_dst + n;                    // n
    float* mbuf  = a_src + n;                    // n (uint menc, then float m, aliased)
    float* den   = mbuf + n;                     // n
    float* ebuf  = den + n;                      // E (logits, then exp(e-m))

    const long total = (long)n * OUT_DIM;

    k_init<<<(int)((total + 255) / 256), 256, 0, stream>>>(out, den, (unsigned*)mbuf, n, total);

    {
        int tilesM = (n + 15) / 16;
        int blocks = (tilesM + 3) / 4;           // 4 waves (M-tiles) per 128-thread block
        k_gemm_wmma<<<blocks, 128, 0, stream>>>(h, W_w, W_b, Wh, n);
    }

    k_node_att<<<(n + 7) / 8, 256, 0, stream>>>(Wh, A_w, a_dst, a_src, n);

    k_edge_logit<<<(E + 255) / 256, 256, 0, stream>>>(src, dst, a_dst, a_src, A_b,
                                                      ebuf, (unsigned*)mbuf, E);

    k_fix_m<<<(n + 255) / 256, 256, 0, stream>>>((unsigned*)mbuf, mbuf, n);

    k_edge_exp<<<(E + 255) / 256, 256, 0, stream>>>(dst, mbuf, ebuf, den, E);

    k_aggregate<<<(E + 7) / 8, 256, 0, stream>>>(src, dst, ebuf, den, Wh, out, E);
}